// LSTM_MVTS_LRN_45543833206927
// MI455X (gfx1250) — compile-verified
//
#include <hip/hip_runtime.h>
#include <hip/hip_bf16.h>

// Problem constants (match the reference).
#define Hdim 33
#define Gdim 132      // 4*H
#define Tlen 100000
#define NPRED 20
#define NTHR 160      // 5 wave32

typedef __attribute__((ext_vector_type(2))) float v2f;
typedef __attribute__((ext_vector_type(8))) float v8f;

// ---------------------------------------------------------------------------
// Kernel 1: x_proj = mvts @ W_ih1^T + (b_ih1 + b_hh1)   -- (100000 x 132)
// One wave per 16x16 output tile, V_WMMA_F32_16X16X4_F32.
// K = 33: iterations k0=0..28 are unconditionally in-bounds (ka+1 <= 31),
// only the k0=32 tail needs masking -> done branchlessly via clamped
// addresses + v_cndmask on the loaded data (no EXEC juggling).
// ---------------------------------------------------------------------------
__global__ __launch_bounds__(32) void xproj_gemm_wmma(
    const float* __restrict__ mvts,   // (T, H)
    const float* __restrict__ Wih1,   // (4H, H)
    const float* __restrict__ bih1,   // (4H)
    const float* __restrict__ bhh1,   // (4H)
    float* __restrict__ xproj)        // (T, 4H)
{
    const int tTile = blockIdx.x;                 // 0..6249 (6250*16 == 100000)
    const int nTile = blockIdx.y;                 // 0..8    (9*16 = 144 >= 132)
    const int lane  = threadIdx.x;                // 0..31
    const int m     = lane & 15;                  // M for A-frag / N for B,C
    const int half  = lane >> 4;                  // K-half selector
    const int rowA  = tTile * 16 + m;
    const int n     = nTile * 16 + m;
    const bool nok  = (n < Gdim);
    const int nc    = nok ? n : (Gdim - 1);       // clamped, always-valid row

    const float* __restrict__ arow = mvts + (long)rowA * Hdim;
    const float* __restrict__ brow = Wih1 + (long)nc   * Hdim;

    v8f acc = {};
    #pragma unroll
    for (int k0 = 0; k0 < 32; k0 += 4) {
        const int ka = k0 + 2 * half;             // ka+1 <= 31 < 33: no guard
        v2f a, b;
        a.x = arow[ka];
        a.y = arow[ka + 1];
        const float b0 = brow[ka];
        const float b1 = brow[ka + 1];
        b.x = nok ? b0 : 0.0f;                    // v_cndmask, branchless
        b.y = nok ? b1 : 0.0f;
        acc = __builtin_amdgcn_wmma_f32_16x16x4_f32(
            false, a, false, b, (short)0, acc, false, false);
    }
    {   // K tail: only K=32 is real, and only for half==0 lanes.
        const int ka = half ? 0 : 32;             // clamped safe address
        const float av = arow[ka];
        const float bv = brow[ka];
        v2f a, b;
        a.x = half ? 0.0f : av;
        a.y = 0.0f;
        b.x = (half || !nok) ? 0.0f : bv;
        b.y = 0.0f;
        acc = __builtin_amdgcn_wmma_f32_16x16x4_f32(
            false, a, false, b, (short)0, acc, false, false);
    }

    if (nok) {
        const float bias = bih1[n] + bhh1[n];
        #pragma unroll
        for (int r = 0; r < 8; ++r) {
            const int mrow = tTile * 16 + r + 8 * half;   // C/D layout
            xproj[(long)mrow * Gdim + n] = acc[r] + bias;
        }
    }
}

// ---------------------------------------------------------------------------
// Fast device nonlinearities (transcendental VALU: v_exp_f32 + v_rcp_f32).
// ---------------------------------------------------------------------------
__device__ __forceinline__ float fsig(float x) {
    return __builtin_amdgcn_rcpf(1.0f + __expf(-x));
}
__device__ __forceinline__ float ftanh(float x) {
    return __builtin_fmaf(2.0f, fsig(2.0f * x), -1.0f);
}

// ---------------------------------------------------------------------------
// Kernel 2: 100000-step serial LSTM recurrence + 20 decode steps.
// One persistent workgroup (160 thr = 5 wave32). Thread j owns gate row
// jc=min(j,131): W_hh1[jc][*] pinned in 33 VGPRs; h/c/gates in LDS.
// Threads 132..159 compute harmless duplicates into gs padding — the hot
// gate phase is completely branch-free. Two barriers per step. The next
// x_proj row is loaded FIRST each iteration so its L2 latency hides behind
// the dot product + barrier + elementwise phase.
// ---------------------------------------------------------------------------
__global__ __launch_bounds__(NTHR) void lstm_scan_decode(
    const float* __restrict__ xproj,  // (T, 4H)
    const float* __restrict__ Whh1,   // (4H, H)
    const float* __restrict__ Wih2,   // (4H, H)
    const float* __restrict__ bih2,   // (4H)
    const float* __restrict__ bhh2,   // (4H)
    float* __restrict__ out)          // (NPRED, H)
{
    __shared__ float hs[Hdim];
    __shared__ float cs[Hdim];
    __shared__ float gs[NTHR];        // 132 real gates + padding slots

    const int j  = threadIdx.x;
    const int jc = (j < Gdim) ? j : (Gdim - 1);   // clamped gate row

    // Pin this thread's recurrent weight row in VGPRs (dupes for j>=132).
    float w1[Hdim];
    #pragma unroll
    for (int k = 0; k < Hdim; ++k) w1[k] = Whh1[jc * Hdim + k];

    if (j < Hdim) { hs[j] = 0.0f; cs[j] = 0.0f; }

    float xp = xproj[jc];                          // pipelined x_proj row, t=0
    __syncthreads();

    for (int t = 0; t < Tlen; ++t) {
        // Issue next row's load immediately (clamped index, branchless).
        const int tn = (t + 1 < Tlen) ? (t + 1) : t;
        const float xnext = xproj[(long)tn * Gdim + jc];

        // 33-MAC dot, 4 independent accumulators to break the FMA chain.
        float a0 = 0.f, a1 = 0.f, a2 = 0.f, a3 = 0.f;
        #pragma unroll
        for (int k = 0; k < 32; k += 4) {
            a0 = __builtin_fmaf(w1[k + 0], hs[k + 0], a0);
            a1 = __builtin_fmaf(w1[k + 1], hs[k + 1], a1);
            a2 = __builtin_fmaf(w1[k + 2], hs[k + 2], a2);
            a3 = __builtin_fmaf(w1[k + 3], hs[k + 3], a3);
        }
        a0 = __builtin_fmaf(w1[32], hs[32], a0);
        gs[j] = xp + ((a0 + a1) + (a2 + a3));      // j>=132 hits padding
        __syncthreads();

        if (j < Hdim) {
            const float ig = fsig(gs[j]);
            const float fg = fsig(gs[j + 33]);
            const float gg = ftanh(gs[j + 66]);
            const float og = fsig(gs[j + 99]);
            const float cn = __builtin_fmaf(fg, cs[j], ig * gg);
            cs[j] = cn;
            hs[j] = og * ftanh(cn);
        }
        xp = xnext;
        __syncthreads();
    }

    // ---- decode: 20 serial steps, x_{s+1} = h(x_s); the reference carries
    // no c/h state here, so the f-gate and W_hh2 drop out exactly.
    float w2[Hdim];
    #pragma unroll
    for (int k = 0; k < Hdim; ++k) w2[k] = Wih2[jc * Hdim + k];
    const float b2j = bih2[jc] + bhh2[jc];
    __syncthreads();

    for (int s = 0; s < NPRED; ++s) {
        float a0 = 0.f, a1 = 0.f, a2 = 0.f, a3 = 0.f;
        #pragma unroll
        for (int k = 0; k < 32; k += 4) {
            a0 = __builtin_fmaf(w2[k + 0], hs[k + 0], a0);
            a1 = __builtin_fmaf(w2[k + 1], hs[k + 1], a1);
            a2 = __builtin_fmaf(w2[k + 2], hs[k + 2], a2);
            a3 = __builtin_fmaf(w2[k + 3], hs[k + 3], a3);
        }
        a0 = __builtin_fmaf(w2[32], hs[32], a0);
        gs[j] = b2j + ((a0 + a1) + (a2 + a3));
        __syncthreads();

        if (j < Hdim) {
            const float ig = fsig(gs[j]);
            const float gg = ftanh(gs[j + 66]);
            const float og = fsig(gs[j + 99]);
            const float hn = og * ftanh(ig * gg);
            hs[j] = hn;
            out[s * Hdim + j] = hn;
        }
        __syncthreads();
    }
}

// ---------------------------------------------------------------------------
// Launch: WMMA GEMM fills x_proj in d_ws (52.8 MB, resident in 192 MB L2),
// then one persistent block runs the latency-bound recurrence + decode.
// ---------------------------------------------------------------------------
extern "C" void kernel_launch(void* const* d_in, const int* in_sizes, int n_in,
                              void* d_out, int out_size, void* d_ws, size_t ws_size,
                              hipStream_t stream) {
    (void)in_sizes; (void)n_in; (void)out_size; (void)ws_size;

    const float* mvts = (const float*)d_in[0];
    const float* Wih1 = (const float*)d_in[1];
    const float* Whh1 = (const float*)d_in[2];
    const float* bih1 = (const float*)d_in[3];
    const float* bhh1 = (const float*)d_in[4];
    const float* Wih2 = (const float*)d_in[5];
    /* d_in[6] = W_hh2 : mathematically unused by the reference decode */
    const float* bih2 = (const float*)d_in[7];
    const float* bhh2 = (const float*)d_in[8];
    float* out   = (float*)d_out;
    float* xproj = (float*)d_ws;           // needs 100000*132*4 = 52.8 MB

    dim3 grid(Tlen / 16, (Gdim + 15) / 16);   // 6250 x 9 tiles, one wave each
    xproj_gemm_wmma<<<grid, 32, 0, stream>>>(mvts, Wih1, bih1, bhh1, xproj);
    lstm_scan_decode<<<1, NTHR, 0, stream>>>(xproj, Whh1, Wih2, bih2, bhh2, out);
}